// MSMDL_DDI_51161650430430
// MI455X (gfx1250) — compile-verified
//
#include <hip/hip_runtime.h>
#include <math.h>
#include <stddef.h>

// ---- static dims from the reference ----
#define NN    32768      // nodes per side
#define BGR   1024       // graphs per side
#define NPG   32         // atoms per graph
#define EIN   131072     // intra directed edges per side
#define HF    128        // hidden dim
#define PROJ  64         // GAT heads*dim
#define INF0  70         // input feature dim
#define INP0  72         // padded to multiple of 4 for f32 WMMA (K=4 steps)
#define LB    3
#define KKDIM 128

typedef float v2f __attribute__((ext_vector_type(2)));
typedef float v8f __attribute__((ext_vector_type(8)));

__device__ __forceinline__ float wsum(float v){
#pragma unroll
  for (int m = 16; m; m >>= 1) v += __shfl_xor(v, m, 32);
  return v;
}
__device__ __forceinline__ float wmaxr(float v){
#pragma unroll
  for (int m = 16; m; m >>= 1) v = fmaxf(v, __shfl_xor(v, m, 32));
  return v;
}

// ---------------- generic fill ----------------
__global__ void k_fill(float* p, int n, float v){
  int i = blockIdx.x * blockDim.x + threadIdx.x;
  if (i < n) p[i] = v;
}

// ---------------- pad block-0 lin0 weight [70,128] -> [72,128] zero rows 70..71 ----------------
__global__ void k_padw(const float* __restrict__ w, float* __restrict__ wp){
  int i = blockIdx.x * blockDim.x + threadIdx.x;
  if (i >= INP0 * HF) return;
  int r = i >> 7;                       // /128
  wp[i] = (r < INF0) ? w[i] : 0.f;
}

// ---------------- LayerNorm (per node over F feats), optional ELU, pad to FP ----------------
__global__ void k_layernorm(const float* __restrict__ x, float* __restrict__ y,
                            const float* __restrict__ g, const float* __restrict__ b,
                            int F, int FP, int do_elu){
  int wid  = (blockIdx.x * blockDim.x + threadIdx.x) >> 5;
  int lane = threadIdx.x & 31;
  if (wid >= NN) return;
  const float* xr = x + (size_t)wid * F;
  float s = 0.f, s2 = 0.f;
  for (int c = lane; c < F; c += 32){ float v = xr[c]; s += v; s2 += v*v; }
  s = wsum(s); s2 = wsum(s2);
  float mu   = s / (float)F;
  float var  = s2 / (float)F - mu*mu;
  float rstd = rsqrtf(var + 1e-5f);
  float* yr = y + (size_t)wid * FP;
  for (int c = lane; c < FP; c += 32){
    float o = 0.f;
    if (c < F){
      o = (xr[c] - mu) * rstd * g[c] + b[c];
      if (do_elu) o = (o > 0.f) ? o : (expf(o) - 1.f);
    }
    yr[c] = o;
  }
}

// ---------------- WMMA dual GEMM:  OUT = act(A1@W1 + b1 [+ (A2*mask)@W2 + b2]) ----------------
// Compile-time K4/OC => fully unrolled, branch-free inner loop (ds_load_b64 + 2x b32 + wmma).
// f32 WMMA 16x16x4: A frag 2 VGPRs (M=lane&15; K pair by lane>>4),
// B frag 2 VGPRs (col=lane&15; K pair by lane>>4), C/D: VGPR r -> row r + 8*(lane>>4).
template<int K4, int OC>
__global__ __launch_bounds__(256) void k_gemm(
    const float* __restrict__ A1, const float* __restrict__ W1, const float* __restrict__ b1,
    const float* __restrict__ A2, const float* __restrict__ W2, const float* __restrict__ b2,
    const float* __restrict__ mask2, float* __restrict__ OUT, int act){
  __shared__ float lA[2][32][K4 + 4];
  const int tid  = threadIdx.x;
  const int lane = tid & 31, wv = tid >> 5;
  const int hsel = lane >> 4, l16 = lane & 15;
  const int row0 = blockIdx.x * 32;
  const int hasA2 = (A2 != nullptr);
  // prefetch weights into cache while A tiles stage (global_prefetch_b8)
  for (int i = tid; i < (K4 * OC) / 16; i += 256){
    __builtin_prefetch((const char*)W1 + (size_t)i * 64, 0, 1);
    if (hasA2) __builtin_prefetch((const char*)W2 + (size_t)i * 64, 0, 1);
  }
  for (int i = tid; i < 32 * K4; i += 256){
    int r = i / K4, c = i - r * K4;
    lA[0][r][c] = A1[(size_t)(row0 + r) * K4 + c];
    if (hasA2){
      float v = A2[(size_t)(row0 + r) * K4 + c];
      if (mask2) v *= mask2[row0 + r];
      lA[1][r][c] = v;
    }
  }
  __syncthreads();
  const int tiles = (OC >> 4) * 2;
  for (int t = wv; t < tiles; t += 8){
    int strip = t & 1;
    int ct    = t >> 1;
    int col   = ct * 16 + l16;
    v8f acc = {0.f,0.f,0.f,0.f,0.f,0.f,0.f,0.f};
#pragma unroll
    for (int k = 0; k < K4; k += 4){
      int kk = k + 2 * hsel;
      v2f a;  a.x = lA[0][strip*16 + l16][kk];  a.y = lA[0][strip*16 + l16][kk + 1];
      v2f bb; bb.x = W1[(size_t)kk * OC + col]; bb.y = W1[(size_t)(kk + 1) * OC + col];
      acc = __builtin_amdgcn_wmma_f32_16x16x4_f32(false, a, false, bb, (short)0, acc, false, false);
    }
    if (hasA2){
#pragma unroll
      for (int k = 0; k < K4; k += 4){
        int kk = k + 2 * hsel;
        v2f a;  a.x = lA[1][strip*16 + l16][kk];  a.y = lA[1][strip*16 + l16][kk + 1];
        v2f bb; bb.x = W2[(size_t)kk * OC + col]; bb.y = W2[(size_t)(kk + 1) * OC + col];
        acc = __builtin_amdgcn_wmma_f32_16x16x4_f32(false, a, false, bb, (short)0, acc, false, false);
      }
    }
    float ba = 0.f;
    if (b1) ba += b1[col];
    if (b2) ba += b2[col];
#pragma unroll
    for (int r = 0; r < 8; r++){
      int orow = row0 + strip*16 + hsel*8 + r;
      float v = acc[r] + ba;
      if (act == 1) v = fmaxf(v, 0.f);
      OUT[(size_t)orow * OC + col] = v;
    }
  }
}

// ---------------- edge feature scatter: aggr[dst] += x[src] * mask[src] ----------------
__global__ void k_scatter_feat(const float* __restrict__ x, const int* __restrict__ src,
                               const int* __restrict__ dst, const float* __restrict__ mask,
                               float* __restrict__ aggr, int E){
  int e    = (blockIdx.x * blockDim.x + threadIdx.x) >> 5;
  int lane = threadIdx.x & 31;
  if (e >= E) return;
  int s = src[e], d = dst[e];
  float m = mask ? mask[s] : 1.f;
  const float* xr = x + (size_t)s * HF;
  float* ar = aggr + (size_t)d * HF;
#pragma unroll
  for (int i = 0; i < 4; i++){
    int c = lane + i * 32;
    atomicAdd(&ar[c], xr[c] * m);
  }
}

// ---------------- per-node dot with 128-vec (+bias, opt sigmoid) ----------------
__global__ void k_dot(const float* __restrict__ x, const float* __restrict__ w,
                      const float* __restrict__ bias, float* __restrict__ out, int act){
  int n    = (blockIdx.x * blockDim.x + threadIdx.x) >> 5;
  int lane = threadIdx.x & 31;
  if (n >= NN) return;
  const float* xr = x + (size_t)n * HF;
  float s = 0.f;
#pragma unroll
  for (int i = 0; i < 4; i++){ int c = lane + i * 32; s += xr[c] * w[c]; }
  s = wsum(s);
  if (lane == 0){
    if (bias) s += bias[0];
    if (act == 1) s = 1.f / (1.f + expf(-s));
    out[n] = s;
  }
}

// ---------------- GAT attention scalars per node/head from [N,64] transform ----------------
__global__ void k_att(const float* __restrict__ Pm, const float* __restrict__ attS,
                      const float* __restrict__ attD, float* __restrict__ as_,
                      float* __restrict__ ad_){
  int n    = (blockIdx.x * blockDim.x + threadIdx.x) >> 5;
  int lane = threadIdx.x & 31;
  if (n >= NN) return;
  const float* pr = Pm + (size_t)n * PROJ;
  float a0 = pr[lane]      * attS[lane];
  float a1 = pr[32 + lane] * attS[32 + lane];
  float d0 = pr[lane]      * attD[lane];
  float d1 = pr[32 + lane] * attD[32 + lane];
  a0 = wsum(a0); a1 = wsum(a1); d0 = wsum(d0); d1 = wsum(d1);
  if (lane == 0){
    as_[n*2] = a0; as_[n*2+1] = a1;
    ad_[n*2] = d0; ad_[n*2+1] = d1;
  }
}

// ---------------- intra GAT: edge exp + denom (edges + self loops appended) ----------------
__global__ void k_edge_exp(const float* __restrict__ as_, const float* __restrict__ ad_,
                           const int* __restrict__ src, const int* __restrict__ dst,
                           float* __restrict__ eexp, float* __restrict__ denom, int E){
  int i = blockIdx.x * blockDim.x + threadIdx.x;
  int tot = (E + NN) * 2;
  if (i >= tot) return;
  int e = i >> 1, h = i & 1;
  int s, d;
  if (e < E){ s = src[e]; d = dst[e]; } else { s = d = e - E; }
  float l = as_[s*2 + h] + ad_[d*2 + h];
  l = (l > 0.f) ? l : 0.2f * l;        // leaky 0.2
  float w = expf(l);
  eexp[i] = w;
  atomicAdd(&denom[d*2 + h], w);
}

// rep[:, :64] = intra bias, rep[:, 64:] = 0 (overwritten by dense inter kernel)
__global__ void k_rep_init(float* __restrict__ rep, const float* __restrict__ bias_i){
  int i = blockIdx.x * blockDim.x + threadIdx.x;
  if (i >= NN * HF) return;
  int c = i & 127;
  rep[i] = (c < 64) ? bias_i[c] : 0.f;
}

__global__ void k_intra_agg(const float* __restrict__ Pm, const float* __restrict__ eexp,
                            const float* __restrict__ denom, const int* __restrict__ src,
                            const int* __restrict__ dst, float* __restrict__ rep, int E){
  int e    = (blockIdx.x * blockDim.x + threadIdx.x) >> 5;
  int lane = threadIdx.x & 31;
  if (e >= E + NN) return;
  int s, d;
  if (e < E){ s = src[e]; d = dst[e]; } else { s = d = e - E; }
#pragma unroll
  for (int h = 0; h < 2; h++){
    float alpha = eexp[e*2 + h] / (denom[d*2 + h] + 1e-16f);
    atomicAdd(&rep[(size_t)d * HF + h*32 + lane], Pm[(size_t)s * PROJ + h*32 + lane] * alpha);
  }
}

// ---------------- inter GAT: dense 32x32 attention per pair, WMMA in LDS ----------------
// self loop edge (n,n) duplicates bipartite source j -> fold into doubled diagonal.
__global__ __launch_bounds__(64) void k_inter_dense(
    const float* __restrict__ Psrc, const float* __restrict__ as_src,
    const float* __restrict__ ad_dst, const float* __restrict__ bias_e,
    float* __restrict__ rep_dst){
  __shared__ float Vs[32][68];       // src-transformed values, both heads
  __shared__ float sAs[64];          // a_src per (node, head)
  __shared__ float sAd[64];
  __shared__ float Am[2][32][36];    // normalized attention per head
  const int pair = blockIdx.x;
  const int tid  = threadIdx.x;
  const int lane = tid & 31, head = tid >> 5;
  for (int i = tid; i < 32 * 64; i += 64)
    Vs[i >> 6][i & 63] = Psrc[(size_t)(pair*32 + (i >> 6)) * PROJ + (i & 63)];
  sAs[tid] = as_src[pair*64 + tid];
  sAd[tid] = ad_dst[pair*64 + tid];
  __syncthreads();
  { // softmax row j = lane for this head, over 32 sources + self(=source j)
    int j = lane;
    float adv = sAd[j*2 + head];
    float den = 0.f;
    for (int s = 0; s < 32; s++){
      float l = sAs[s*2 + head] + adv;
      l = (l > 0.f) ? l : 0.2f * l;
      float w = expf(l);
      Am[head][j][s] = w;
      den += w;
    }
    den += Am[head][j][j];                 // self-loop duplicate
    float inv = 1.f / (den + 1e-16f);
    for (int s = 0; s < 32; s++) Am[head][j][s] *= inv;
    Am[head][j][j] *= 2.f;
  }
  __syncthreads();
  const int hsel = lane >> 4, l16 = lane & 15;
#pragma unroll
  for (int ti = 0; ti < 2; ti++){
#pragma unroll
    for (int tj = 0; tj < 2; tj++){
      v8f acc = {0.f,0.f,0.f,0.f,0.f,0.f,0.f,0.f};
#pragma unroll
      for (int k = 0; k < 32; k += 4){
        int kk = k + 2 * hsel;
        v2f a;  a.x = Am[head][ti*16 + l16][kk];  a.y = Am[head][ti*16 + l16][kk + 1];
        v2f bb; bb.x = Vs[kk][head*32 + tj*16 + l16]; bb.y = Vs[kk + 1][head*32 + tj*16 + l16];
        acc = __builtin_amdgcn_wmma_f32_16x16x4_f32(false, a, false, bb, (short)0, acc, false, false);
      }
      int c = head*32 + tj*16 + l16;
      float bia = bias_e[c];
#pragma unroll
      for (int r = 0; r < 8; r++){
        int orow = pair*32 + ti*16 + hsel*8 + r;
        rep_dst[(size_t)orow * HF + 64 + c] = acc[r] + bia;
      }
    }
  }
}

// ---------------- SAG pool ----------------
__global__ void k_score_init(const float* q, const float* brel, float* spre){
  int n = blockIdx.x * blockDim.x + threadIdx.x;
  if (n < NN) spre[n] = q[n] + brel[0];
}
__global__ void k_scatter_scalar(const float* p, const int* __restrict__ src,
                                 const int* __restrict__ dst, float* spre, int E){
  int e = blockIdx.x * blockDim.x + threadIdx.x;
  if (e < E) atomicAdd(&spre[dst[e]], p[src[e]]);
}
__global__ void k_pool_softmax(const float* spre, float* ssm){
  int g    = (blockIdx.x * blockDim.x + threadIdx.x) >> 5;
  int lane = threadIdx.x & 31;
  if (g >= BGR) return;
  float v = spre[g*32 + lane];
  float m = wmaxr(v);
  float e = expf(v - m);
  float s = wsum(e);
  ssm[g*32 + lane] = e / (s + 1e-16f);
}
__global__ __launch_bounds__(128) void k_pool_emb(const float* __restrict__ rep,
                                                  const float* __restrict__ ssm,
                                                  float* __restrict__ emb, int blk){
  int g = blockIdx.x, c = threadIdx.x;
  float acc = 0.f;
  for (int j = 0; j < 32; j++)
    acc += ssm[g*32 + j] * rep[(size_t)(g*32 + j) * HF + c];
  emb[(size_t)g * (LB * HF) + blk * HF + c] = acc;
}

// ---------------- co-attention + RESCAL, one workgroup per pair ----------------
__global__ __launch_bounds__(128) void k_final(
    const float* __restrict__ heads, const float* __restrict__ tails,
    const int* __restrict__ rels, const float* __restrict__ emb,
    const float* __restrict__ wq, const float* __restrict__ wk,
    const float* __restrict__ cbias, const float* __restrict__ avec,
    float* __restrict__ out){
  __shared__ float red[128];
  __shared__ float tmpM[3][128];
  __shared__ float keys[3][64];
  __shared__ float qrs[3][64];
  __shared__ float sc[9], at[9];
  __shared__ float scl[7];
  const int b = blockIdx.x, t = threadIdx.x;
  const float* hb = heads + (size_t)b * 384;
  const float* tb = tails + (size_t)b * 384;
  const float* Em = emb + (size_t)rels[b] * (KKDIM * KKDIM);
  auto reduce = [&](float v)->float{
    red[t] = v; __syncthreads();
    for (int s = 64; s >= 1; s >>= 1){ if (t < s) red[t] += red[t + s]; __syncthreads(); }
    float r = red[0]; __syncthreads();
    return r;
  };
  float v = 0.f;
  for (int i = 0; i < 128; i++){ float d = Em[i*128 + t]; v += d*d; }
  float S = reduce(v);
  if (t == 0) scl[0] = 1.f / (sqrtf(S) + 1e-12f);
  for (int i = 0; i < 3; i++){
    float hv = hb[i*128 + t]; float r = reduce(hv*hv);
    if (t == 0) scl[1 + i] = 1.f / (sqrtf(r) + 1e-12f);
    float tv = tb[i*128 + t]; r = reduce(tv*tv);
    if (t == 0) scl[4 + i] = 1.f / (sqrtf(r) + 1e-12f);
  }
  __syncthreads();
  for (int i = 0; i < 3; i++){
    float acc = 0.f;
    for (int k = 0; k < 128; k++) acc += hb[i*128 + k] * Em[k*128 + t];
    tmpM[i][t] = acc * scl[1 + i];
  }
  if (t < 64){
    for (int i = 0; i < 3; i++){
      float ka = 0.f, qa = 0.f;
      for (int k = 0; k < 128; k++){
        ka += hb[i*128 + k] * wk[k*64 + t];
        qa += tb[i*128 + k] * wq[k*64 + t];
      }
      keys[i][t] = ka; qrs[i][t] = qa;
    }
  }
  __syncthreads();
  if (t < 9){
    int i = t / 3, j = t % 3;
    float s = 0.f;
    for (int m = 0; m < 128; m++) s += tmpM[i][m] * tb[j*128 + m];
    sc[t] = s * scl[0] * scl[4 + j];
    float a = 0.f;
    for (int d = 0; d < 64; d++) a += tanhf(qrs[j][d] + keys[i][d] + cbias[d]) * avec[d];
    at[t] = a;
  }
  __syncthreads();
  if (t == 0){
    float o = 0.f;
    for (int i = 0; i < 9; i++) o += sc[i] * at[i];
    out[b] = o;
  }
}

// =====================================================================================
extern "C" void kernel_launch(void* const* d_in, const int* in_sizes, int n_in,
                              void* d_out, int out_size, void* d_ws, size_t ws_size,
                              hipStream_t stream){
  (void)in_sizes; (void)n_in; (void)out_size; (void)ws_size;
  const float* in_hx = (const float*)d_in[0];
  const float* in_tx = (const float*)d_in[1];
  const int* hsrc = (const int*)d_in[2]; const int* hdst = hsrc + EIN;
  const int* tsrc = (const int*)d_in[3]; const int* tdst = tsrc + EIN;
  // d_in[4] = b_edge_index: unused (dense 32x32 per pair exploited analytically)
  const int* rels = (const int*)d_in[5];
  // d_in[6] = batch: unused (nodes are contiguous per graph)

  // params flattened in JAX pytree order (dict keys alphabetical, lists in order):
  // blocks[3]{convs[3]{lin1{b,w},lin2{b,w}}, inter{att_dst,att_src,bias,w_dst,w_src},
  //           intra{att_dst,att_src,bias,w}, lin0{b,w},
  //           masks[3]{lin1{b,w},lin2{b,w},lin3{b,w}}, pool{b_rel,w_rel,w_root}},
  // coatt{a,bias,w_k,w_q}, net_norms[3]{b,g}, norm0{b,g}, rescal{emb}
  const int PB = 7;
  auto P = [&](int i)->const float*{ return (const float*)d_in[i]; };
  const int I_COATT = PB + 3*44;      // 139
  const int I_NN    = I_COATT + 4;    // 143
  const int I_NORM0 = I_NN + 6;       // 149
  const int I_RESC  = I_NORM0 + 2;    // 151

  float* ws = (float*)d_ws;
  size_t off = 0;
  auto alloc = [&](size_t n){ float* r = ws + off; off += n; return r; };
  float* x0[2]  = { alloc((size_t)NN*INP0), alloc((size_t)NN*INP0) };
  float* fA[2]  = { alloc((size_t)NN*HF),   alloc((size_t)NN*HF) };
  float* fB[2]  = { alloc((size_t)NN*HF),   alloc((size_t)NN*HF) };
  float* aggr   =   alloc((size_t)NN*HF);
  float* rep[2] = { alloc((size_t)NN*HF),   alloc((size_t)NN*HF) };
  float* Pi[2]  = { alloc((size_t)NN*PROJ), alloc((size_t)NN*PROJ) };
  float* Psx[2] = { alloc((size_t)NN*PROJ), alloc((size_t)NN*PROJ) };
  float* Pdx[2] = { alloc((size_t)NN*PROJ), alloc((size_t)NN*PROJ) };
  float* msk[2] = { alloc(NN), alloc(NN) };
  float* asi[2] = { alloc(2*NN), alloc(2*NN) };
  float* adi[2] = { alloc(2*NN), alloc(2*NN) };
  float* asx[2] = { alloc(2*NN), alloc(2*NN) };
  float* adx[2] = { alloc(2*NN), alloc(2*NN) };
  float* dumm   =   alloc(2*NN);
  float* eexp   =   alloc((size_t)(EIN+NN)*2);
  float* denom  =   alloc(2*NN);
  float* pb     =   alloc(NN);
  float* qb     =   alloc(NN);
  float* spre   =   alloc(NN);
  float* ssm    =   alloc(NN);
  float* headsB =   alloc((size_t)BGR*LB*HF);
  float* tailsB =   alloc((size_t)BGR*LB*HF);
  float* Wpad   =   alloc((size_t)INP0*HF);   // zero-padded block-0 lin0 weight

  const int* srcA[2] = { hsrc, tsrc };
  const int* dstA[2] = { hdst, tdst };
  const int GW_NODE = 4096;                    // NN waves / 8 per block
  const int GB_E32  = (EIN*32 + 255)/256;
  const int GB_EA32 = ((EIN+NN)*32 + 255)/256;

  // pad block-0 lin0 weight once per call (makes every GEMM guard-free)
  k_padw<<<(INP0*HF + 255)/256,256,0,stream>>>(P(PB + 22), Wpad);

  // input LayerNorm (70 feats, pad to 72 with zeros)
  k_layernorm<<<GW_NODE,256,0,stream>>>(in_hx, x0[0], P(I_NORM0+1), P(I_NORM0), INF0, INP0, 0);
  k_layernorm<<<GW_NODE,256,0,stream>>>(in_tx, x0[1], P(I_NORM0+1), P(I_NORM0), INF0, INP0, 0);

  float* cur[2] = { fA[0], fA[1] };
  float* alt[2] = { fB[0], fB[1] };
  const float* xin[2] = { x0[0], x0[1] };

  for (int blk = 0; blk < 3; blk++){
    int pbase = PB + blk*44;
    const float* lin0b = P(pbase+21); const float* lin0w = P(pbase+22);
    for (int s = 0; s < 2; s++){
      if (blk == 0)
        k_gemm<INP0,HF><<<NN/32,256,0,stream>>>(xin[s], Wpad, lin0b,
                                                nullptr,nullptr,nullptr,nullptr, cur[s], 0);
      else
        k_gemm<HF,HF><<<NN/32,256,0,stream>>>(xin[s], lin0w, lin0b,
                                              nullptr,nullptr,nullptr,nullptr, cur[s], 0);
    }
    // --- conv iterations ---
    for (int j = 0; j < 3; j++){
      const float* m1b = P(pbase+23+j*6+0); const float* m1w = P(pbase+23+j*6+1);
      const float* m2b = P(pbase+23+j*6+2); const float* m2w = P(pbase+23+j*6+3);
      const float* m3b = P(pbase+23+j*6+4); const float* m3w = P(pbase+23+j*6+5);
      const float* c1b = P(pbase+j*4+0);    const float* c1w = P(pbase+j*4+1);
      const float* c2b = P(pbase+j*4+2);    const float* c2w = P(pbase+j*4+3);
      for (int s = 0; s < 2; s++){
        const float* mprev = (j == 0) ? nullptr : msk[s];
        // WeightConv: mask = sigmoid(lin3(relu(lin1(aggr) + lin2(x*m))))
        k_fill<<<(NN*HF+255)/256,256,0,stream>>>(aggr, NN*HF, 0.f);
        k_scatter_feat<<<GB_E32,256,0,stream>>>(cur[s], srcA[s], dstA[s], mprev, aggr, EIN);
        k_gemm<HF,HF><<<NN/32,256,0,stream>>>(aggr, m1w, m1b, cur[s], m2w, m2b, mprev,
                                              alt[s], 1);
        k_dot<<<GW_NODE,256,0,stream>>>(alt[s], m3w, m3b, msk[s], 1);
        // SparseConv: x = relu(lin1(aggr(x*m)) + lin2(x*m))
        k_fill<<<(NN*HF+255)/256,256,0,stream>>>(aggr, NN*HF, 0.f);
        k_scatter_feat<<<GB_E32,256,0,stream>>>(cur[s], srcA[s], dstA[s], msk[s], aggr, EIN);
        k_gemm<HF,HF><<<NN/32,256,0,stream>>>(aggr, c1w, c1b, cur[s], c2w, c2b, msk[s],
                                              alt[s], 1);
        float* tmp = cur[s]; cur[s] = alt[s]; alt[s] = tmp;
      }
    }
    // --- GAT (elu(relu(x)) == relu(x), so `cur` is already he/te) ---
    const float* i_attd = P(pbase+17); const float* i_atts = P(pbase+18);
    const float* i_bias = P(pbase+19); const float* i_w    = P(pbase+20);
    const float* e_attd = P(pbase+12); const float* e_atts = P(pbase+13);
    const float* e_bias = P(pbase+14); const float* e_wd   = P(pbase+15);
    const float* e_wsrc = P(pbase+16);
    for (int s = 0; s < 2; s++){
      k_gemm<HF,PROJ><<<NN/32,256,0,stream>>>(cur[s], i_w,    nullptr, nullptr,nullptr,nullptr,nullptr, Pi[s],  0);
      k_gemm<HF,PROJ><<<NN/32,256,0,stream>>>(cur[s], e_wsrc, nullptr, nullptr,nullptr,nullptr,nullptr, Psx[s], 0);
      k_gemm<HF,PROJ><<<NN/32,256,0,stream>>>(cur[s], e_wd,   nullptr, nullptr,nullptr,nullptr,nullptr, Pdx[s], 0);
      k_att<<<GW_NODE,256,0,stream>>>(Pi[s],  i_atts, i_attd, asi[s], adi[s]);
      k_att<<<GW_NODE,256,0,stream>>>(Psx[s], e_atts, e_attd, asx[s], dumm);
      k_att<<<GW_NODE,256,0,stream>>>(Pdx[s], e_atts, e_attd, dumm,   adx[s]);
    }
    for (int s = 0; s < 2; s++){
      k_rep_init<<<(NN*HF+255)/256,256,0,stream>>>(rep[s], i_bias);
      k_fill<<<(2*NN+255)/256,256,0,stream>>>(denom, 2*NN, 0.f);
      k_edge_exp<<<((EIN+NN)*2+255)/256,256,0,stream>>>(asi[s], adi[s], srcA[s], dstA[s], eexp, denom, EIN);
      k_intra_agg<<<GB_EA32,256,0,stream>>>(Pi[s], eexp, denom, srcA[s], dstA[s], rep[s], EIN);
    }
    // dense inter attention (writes rep[:,64:128])
    k_inter_dense<<<BGR,64,0,stream>>>(Psx[0], asx[0], adx[1], e_bias, rep[1]); // h -> t
    k_inter_dense<<<BGR,64,0,stream>>>(Psx[1], asx[1], adx[0], e_bias, rep[0]); // t -> h
    // --- SAG pooling -> per-block embeddings ---
    const float* prelb = P(pbase+41); const float* prelw = P(pbase+42); const float* prootw = P(pbase+43);
    float* embB[2] = { headsB, tailsB };
    for (int s = 0; s < 2; s++){
      k_dot<<<GW_NODE,256,0,stream>>>(rep[s], prelw,  nullptr, pb, 0);
      k_dot<<<GW_NODE,256,0,stream>>>(rep[s], prootw, nullptr, qb, 0);
      k_score_init<<<(NN+255)/256,256,0,stream>>>(qb, prelb, spre);
      k_scatter_scalar<<<(EIN+255)/256,256,0,stream>>>(pb, srcA[s], dstA[s], spre, EIN);
      k_pool_softmax<<<(BGR*32+255)/256,256,0,stream>>>(spre, ssm);
      k_pool_emb<<<BGR,128,0,stream>>>(rep[s], ssm, embB[s], blk);
    }
    // --- next block input: elu(LN(rep)) ---
    if (blk < 2){
      const float* nnb = P(I_NN + blk*2); const float* nng = P(I_NN + blk*2 + 1);
      for (int s = 0; s < 2; s++){
        k_layernorm<<<GW_NODE,256,0,stream>>>(rep[s], alt[s], nng, nnb, HF, HF, 1);
        xin[s] = alt[s];
      }
    }
  }
  // --- co-attention + RESCAL ---
  k_final<<<BGR,128,0,stream>>>(headsB, tailsB, rels, P(I_RESC),
                                P(I_COATT+3), P(I_COATT+2), P(I_COATT+1), P(I_COATT+0),
                                (float*)d_out);
}